// ROIPool_90692529423152
// MI455X (gfx1250) — compile-verified
//
#include <hip/hip_runtime.h>
#include <math.h>

// ---------------------------------------------------------------------------
// Problem constants (from the reference setup_inputs / _roi_pool)
// ---------------------------------------------------------------------------
static constexpr int   kB = 4;
static constexpr int   kC = 256;
static constexpr int   kH = 50;
static constexpr int   kW = 50;
static constexpr int   kR = 128;
static constexpr int   kP = 7;
static constexpr float kScale = 0.0625f;

static constexpr int kBinsPerCh = kP * kP;     // 49
static constexpr int kChTile    = 8;           // channels staged per block
static constexpr int kWinSide   = 24;          // >= max window side for this input dist
static constexpr int kWinCap    = kWinSide * kWinSide;  // floats per channel window
static constexpr int kThreads   = 256;         // 8 wave32 waves

// ---------------------------------------------------------------------------
// One block per (roi, 8-channel tile).
//  1. All window geometry is uniform per block (scalar ALU / s_load for ROI).
//  2. Stage each channel's ROI window global->LDS with CDNA5 async-DMA loads
//     (global_load_async_to_lds_b32, tracked by ASYNCcnt) — one coalesced
//     touch per window element instead of overlapping per-bin gathers.
//  3. s_wait_asynccnt 0 (per wave) + __syncthreads(), then compute the 8*49
//     bin maxima from LDS.
//  Fallback (uniform branch) to direct-global gathering if a window ever
//  exceeds the LDS cap, so the kernel is correct for arbitrary inputs.
// Rounding: jnp.round is round-half-even -> rintf (v_rndne_f32).
// Empty bins (clipped away) produce -inf which the reference maps to 0.
// ---------------------------------------------------------------------------
__global__ __launch_bounds__(kThreads)
void roi_pool_async_kernel(const float* __restrict__ feat,
                           const float* __restrict__ rois,
                           float* __restrict__ out,
                           int total) {
    __shared__ float smem[kChTile * kWinCap];   // 18 KB of the 320 KB WGP LDS

    constexpr int tilesPerRoi = kC / kChTile;   // 32
    const int r  = blockIdx.x / tilesPerRoi;                    // uniform
    const int c0 = (blockIdx.x % tilesPerRoi) * kChTile;        // uniform

    // ROI row: address depends only on blockIdx -> scalar loads + scalar float ops
    const float* roi = rois + r * 5;
    const int   bidx = (int)roi[0];
    const float x1   = rintf(roi[1] * kScale);
    const float y1   = rintf(roi[2] * kScale);
    const float x2   = rintf(roi[3] * kScale);
    const float y2   = rintf(roi[4] * kScale);
    const float bw   = fmaxf(x2 - x1 + 1.0f, 1.0f) / (float)kP;
    const float bh   = fmaxf(y2 - y1 + 1.0f, 1.0f) / (float)kP;

    // Full-ROI window on the feature map = union of all 7x7 bin windows.
    const int ws0 = (int)fminf(fmaxf(x1, 0.0f), (float)kW);                       // floor(bw*0)+x1
    const int weL = (int)fminf(fmaxf(ceilf(bw * 7.0f) + x1, 0.0f), (float)kW);
    const int hs0 = (int)fminf(fmaxf(y1, 0.0f), (float)kH);
    const int heL = (int)fminf(fmaxf(ceilf(bh * 7.0f) + y1, 0.0f), (float)kH);
    const int ww    = weL - ws0;
    const int wh    = heL - hs0;
    const int winsz = ww * wh;

    const size_t planeStride = (size_t)(kH * kW);
    const float* base = feat + ((size_t)(bidx * kC + c0)) * planeStride;

    const bool staged = (winsz > 0) && (winsz <= kWinCap);      // uniform per block

    if (staged) {
        // ------- async global -> LDS staging of the 8 channel windows -------
        const int totElems = kChTile * winsz;
        for (int e = threadIdx.x; e < totElems; e += kThreads) {
            const int ch  = e / winsz;
            const int rem = e - ch * winsz;
            const int row = rem / ww;
            const int col = rem - row * ww;
            const float* gp =
                base + (size_t)ch * planeStride + (size_t)(hs0 + row) * kW + (ws0 + col);
            // Low 32 bits of the generic shared-aperture address == LDS byte offset.
            unsigned ldsAddr = (unsigned)(uintptr_t)(&smem[ch * kWinCap + rem]);
            asm volatile("global_load_async_to_lds_b32 %0, %1, off"
                         :: "v"(ldsAddr), "v"(gp)
                         : "memory");
        }
        // Wait for this wave's async DMA into LDS to land (ASYNCcnt -> 0).
        asm volatile("s_wait_asynccnt 0" ::: "memory");
    }
    __syncthreads();  // publish all waves' LDS window writes

    // ----------------------- per-bin max reduction -------------------------
    for (int o = threadIdx.x; o < kChTile * kBinsPerCh; o += kThreads) {
        const int ch  = o / kBinsPerCh;
        const int bin = o - ch * kBinsPerCh;
        const int ph  = bin / kP;
        const int pw  = bin - ph * kP;

        const int ws = (int)fminf(fmaxf(floorf(bw * (float)pw)       + x1, 0.0f), (float)kW);
        const int we = (int)fminf(fmaxf(ceilf (bw * (float)(pw + 1)) + x1, 0.0f), (float)kW);
        const int hs = (int)fminf(fmaxf(floorf(bh * (float)ph)       + y1, 0.0f), (float)kH);
        const int he = (int)fminf(fmaxf(ceilf (bh * (float)(ph + 1)) + y1, 0.0f), (float)kH);

        float m = -INFINITY;
        if (staged) {
            const float* wp = &smem[ch * kWinCap];
            for (int h = hs; h < he; ++h) {
                const float* rowp = wp + (h - hs0) * ww - ws0;   // index by absolute w
                for (int w = ws; w < we; ++w) m = fmaxf(m, rowp[w]);
            }
        } else {
            const float* plane = base + (size_t)ch * planeStride;
            for (int h = hs; h < he; ++h) {
                const float* rowp = plane + h * kW;
                for (int w = ws; w < we; ++w) m = fmaxf(m, rowp[w]);
            }
        }

        const int gidx = (r * kC + (c0 + ch)) * kBinsPerCh + bin;
        if (gidx < total)
            out[gidx] = (m == -INFINITY) ? 0.0f : m;   // empty (clipped) bin -> 0
    }
}

// ---------------------------------------------------------------------------
// Host-side launcher
//   d_in[0] = input feature map, f32, B*C*H*W
//   d_in[1] = rois,               f32, R*5
//   d_out   = f32, R*C*P*P
// ---------------------------------------------------------------------------
extern "C" void kernel_launch(void* const* d_in, const int* in_sizes, int n_in,
                              void* d_out, int out_size, void* d_ws, size_t ws_size,
                              hipStream_t stream) {
    (void)in_sizes; (void)n_in; (void)d_ws; (void)ws_size;

    const float* feat = (const float*)d_in[0];
    const float* rois = (const float*)d_in[1];
    float*       out  = (float*)d_out;

    const int nblocks = kR * (kC / kChTile);   // 128 * 32 = 4096
    roi_pool_async_kernel<<<nblocks, kThreads, 0, stream>>>(feat, rois, out, out_size);
}